// BispectrumCalculator_57741540327690
// MI455X (gfx1250) — compile-verified
//
#include <hip/hip_runtime.h>
#include <hip/hip_bf16.h>

typedef __attribute__((ext_vector_type(2))) float v2f;
typedef __attribute__((ext_vector_type(8))) float v8f;

namespace {
constexpr int kL = 256;   // signal length
constexpr int kT = 8;     // targets per batch
constexpr int kB = 32;    // batch
constexpr int kPad = 2 * kL;          // per-signal padded length (256 zeros + 256 data)
constexpr int kTilesTri = 136;        // 16*17/2 upper-triangle 16x16 tiles
}

// One wave computes one upper-triangle 16x16 tile of the symmetric LxL
// bispectrum for batch b, accumulating over all T signals, then stores the
// tile and (for off-diagonal tiles) its mirror. A-operand row l1:
// a[k] = x[k]*x[k-l1]; B-operand column l2: b[k] = x[k-l2]; x is left
// zero-padded in LDS so no bounds branches are needed.
__global__ __launch_bounds__(256)
void bispectrum_wmma_kernel(const float* __restrict__ target,
                            float* __restrict__ out) {
    __shared__ float xs[kT * kPad];   // per t: [0..255]=0 pad, [256..511]=x

    const int tid   = threadIdx.x;
    const int lane  = tid & 31;
    const int wave  = tid >> 5;
    const int lm    = lane & 15;    // row (A) / col (B) within tile
    const int half  = lane >> 4;    // selects K-pair {0,1} vs {2,3}
    const int b     = blockIdx.y;

    // ---- decode upper-triangle tile index (wave-uniform) ----
    const int w = blockIdx.x * 8 + wave;          // 0..135
    int tile_m = 0, base = 0;
    while (w >= base + (16 - tile_m)) {           // uniform loop, <=16 iters
        base += 16 - tile_m;
        ++tile_m;
    }
    const int tile_n = tile_m + (w - base);       // tile_n >= tile_m
    const int l1 = tile_m * 16 + lm;
    const int l2 = tile_n * 16 + lm;

    // ---- stage all T padded signals into LDS in one pass ----
    const float* xb = target + (size_t)b * (kT * kL);
#pragma unroll
    for (int i = tid; i < kT * kPad; i += 256) {
        const int t = i >> 9;          // i / 512
        const int j = i & (kPad - 1);  // i % 512
        xs[i] = (j < kL) ? 0.0f : xb[t * kL + (j - kL)];
    }
    __syncthreads();

    // ---- accumulate over all T signals and full K=256 ----
    v8f c = {};
    for (int t = 0; t < kT; ++t) {
        const float* pA = &xs[t * kPad + kL - l1];  // pA[k] = x[k - l1]
        const float* pB = &xs[t * kPad + kL - l2];  // pB[k] = x[k - l2]
        const float* pX = &xs[t * kPad + kL];       // pX[k] = x[k]

#pragma unroll 8
        for (int kb = 0; kb < kL; kb += 4) {
            const int k0 = kb + half * 2;
            const float x0 = pX[k0];
            const float x1 = pX[k0 + 1];
            v2f a, bv;
            a.x  = x0 * pA[k0];
            a.y  = x1 * pA[k0 + 1];
            bv.x = pB[k0];
            bv.y = pB[k0 + 1];
            // D = A(16x4) * B(4x16) + C, fp32 throughout
            c = __builtin_amdgcn_wmma_f32_16x16x4_f32(
                    /*neg_a=*/false, a, /*neg_b=*/false, bv,
                    /*c_mod=*/(short)0, c, /*reuse_a=*/false, /*reuse_b=*/false);
        }
    }

    // ---- scale by 1/(L*T) and store tile (+ mirrored tile) ----
    const float sc = 1.0f / (float)(kL * kT);
    float* ob = out + (size_t)b * kL * kL;

    // natural layout: row = tile_m*16 + 8*half + r, col = tile_n*16 + lm
    float* op = ob + (size_t)(tile_m * 16 + 8 * half) * kL + tile_n * 16 + lm;
#pragma unroll
    for (int r = 0; r < 8; ++r) {
        op[r * kL] = c[r] * sc;
    }

    if (tile_m != tile_n) {           // wave-uniform branch
        // mirror: row = tile_n*16 + lm, col = tile_m*16 + 8*half + r
        // -> per lane the 8 elements are consecutive in memory
        float* opT = ob + (size_t)(tile_n * 16 + lm) * kL + tile_m * 16 + 8 * half;
#pragma unroll
        for (int r = 0; r < 8; ++r) {
            opT[r] = c[r] * sc;
        }
    }
}

extern "C" void kernel_launch(void* const* d_in, const int* in_sizes, int n_in,
                              void* d_out, int out_size, void* d_ws, size_t ws_size,
                              hipStream_t stream) {
    const float* target = (const float*)d_in[0];
    float* out = (float*)d_out;

    // 136 upper-triangle tiles per batch, 8 waves/block -> 17 blocks per batch
    dim3 grid(kTilesTri / 8, kB);
    bispectrum_wmma_kernel<<<grid, 256, 0, stream>>>(target, out);

    // second tuple element: pass-through copy of target after source block
    hipMemcpyAsync(out + (size_t)kB * kL * kL, target,
                   (size_t)kB * kT * kL * sizeof(float),
                   hipMemcpyDeviceToDevice, stream);
}